// SupervisedContrastiveLoss_56461640073787
// MI455X (gfx1250) — compile-verified
//
#include <hip/hip_runtime.h>
#include <hip/hip_bf16.h>
#include <math.h>

// ---------------------------------------------------------------------------
// Fused supervised-contrastive loss for MI455X (gfx1250, wave32).
//   B=2048, D=128, 2N=4096, TAU=0.1, EPS=1e-8.
// - 16x16 sim tiles via chained v_wmma_f32_16x16x4_f32 (fp32 fidelity).
// - B panels staged into padded LDS by the Tensor Data Mover
//   (tensor_load_to_lds, TENSORcnt) with double buffering, so the DMA of
//   panel p+1 overlaps the WMMA + exp epilogue of panel p.
// - Never materializes the 4096x4096 sim matrix; deterministic final reduce.
// ---------------------------------------------------------------------------

typedef __attribute__((ext_vector_type(2))) float        v2f;
typedef __attribute__((ext_vector_type(8))) float        v8f;
typedef __attribute__((ext_vector_type(4))) unsigned int v4u;
typedef __attribute__((ext_vector_type(8))) int          v8i;
typedef __attribute__((ext_vector_type(4))) int          v4i;

#define N2    4096
#define NHALF 2048
#define D     128
#define TAU   0.1f
#define EPS   1e-8f

#define BM    128   // rows per workgroup (8 waves x 16)
#define BN    64    // columns per staged panel
#define LDB   132   // padded LDS row stride (floats): 128 data + 4 pad DWORDs
#define NP    (N2 / BN)

#if __has_builtin(__builtin_amdgcn_tensor_load_to_lds) && \
    __has_builtin(__builtin_amdgcn_s_wait_tensorcnt)
#define USE_TDM 1
#else
#define USE_TDM 0
#endif

__device__ __forceinline__ const float* prow(const float* zi, const float* zj, int i) {
  return (i < NHALF) ? (zi + (size_t)i * D) : (zj + (size_t)(i - NHALF) * D);
}

#if USE_TDM
// One TDM descriptor moves a 64x128 f32 tile (rows contiguous, stride 128)
// into LDS with 4 pad DWORDs inserted every 128 DWORDs -> stride LDB=132.
__device__ __forceinline__ void tdm_load_panel(const float* gsrc, void* ldst) {
  unsigned long long ga = (unsigned long long)(size_t)gsrc;   // byte VA [56:0]
  unsigned           la = (unsigned)(size_t)ldst;             // LDS byte addr

  v4u g0;
  g0.x = 1u;                                            // count=1, user D#
  g0.y = la;                                            // lds_addr [63:32]
  g0.z = (unsigned)ga;                                  // global_addr lo
  g0.w = (unsigned)((ga >> 32) & 0x01FFFFFFu)           // global_addr [56:32]
         | (2u << 30);                                  // type=2 ("image")

  v8i g1;
  g1[0] = (2 << 16)      // data_size = 4 bytes
        | (1 << 20)      // pad_enable
        | (6 << 22)      // pad_interval: 2^(6+1) = 128 DWORDs
        | (3 << 25);     // pad_amount:  3+1     = 4 DWORDs
  g1[1] = (int)(128u << 16);   // tensor_dim0 = 128 (lo16 in bits 63:48)
  g1[2] = (int)(2048u << 16);  // tensor_dim0 hi=0 | tensor_dim1 lo16 = 2048
  g1[3] = (int)(128u << 16);   // tensor_dim1 hi=0 | tile_dim0 = 128
  g1[4] = 64;                  // tile_dim1 = 64 | tile_dim2 = 0
  g1[5] = 128;                 // tensor_dim0_stride lo32 = 128 elements
  g1[6] = 0;                   // stride hi | tensor_dim1_stride lo16
  g1[7] = 0;                   // tensor_dim1_stride hi

  v4i z4 = {0, 0, 0, 0};               // groups 2/3 unused (2-D tile)
  v8i z8 = {0, 0, 0, 0, 0, 0, 0, 0};   // trailing descriptor group (6-arg form)
  __builtin_amdgcn_tensor_load_to_lds(g0, g1, z4, z4, z8, 0);
}
#endif

// -------------------------------- kernel 1: row norms ----------------------
__global__ void scl_norms_kernel(const float* __restrict__ zi,
                                 const float* __restrict__ zj,
                                 float* __restrict__ norms) {
  int gid  = blockIdx.x * blockDim.x + threadIdx.x;
  int row  = gid >> 5;          // one wave32 per row
  int lane = threadIdx.x & 31;
  if (row >= N2) return;
  const float* r = prow(zi, zj, row);
  float4 v = *(const float4*)(r + lane * 4);
  float s = v.x * v.x + v.y * v.y + v.z * v.z + v.w * v.w;
  s += __shfl_xor(s, 16, 32);
  s += __shfl_xor(s, 8, 32);
  s += __shfl_xor(s, 4, 32);
  s += __shfl_xor(s, 2, 32);
  s += __shfl_xor(s, 1, 32);
  if (lane == 0) norms[row] = sqrtf(s);
}

// ------------------------- kernel 2: fused sim + row stats -----------------
__global__ void __launch_bounds__(256)
scl_main_kernel(const float* __restrict__ zi,
                const float* __restrict__ zj,
                const int*   __restrict__ y,
                const float* __restrict__ norms,
                float* __restrict__ mlp /* per-row mean log-prob of positives */) {
  __shared__ float Bs[2][BN * LDB];
  __shared__ float nB[2][BN];
  __shared__ int   yB[2][BN];

  const int tid  = threadIdx.x;
  const int lane = tid & 31;
  const int lo   = lane & 15;        // N within tile / M within half
  const int hi   = lane >> 4;        // 0: lanes 0-15, 1: lanes 16-31
  const int hi8  = hi << 3;
  const int kOff = hi << 1;          // K sub-offset {0,2} per ISA layout

  const int rowBase = blockIdx.x * BM + (tid >> 5) * 16;

  // --- A fragments for this wave's 16 rows, all K, kept in registers -------
  // lane<16: A[M=lo][4k+{0,1}] ; lane>=16: A[M=lo][4k+{2,3}]
  const float* pa = prow(zi, zj, rowBase + lo);
  v2f areg[32];
#pragma unroll
  for (int ks = 0; ks < 32; ++ks)
    areg[ks] = *(const v2f*)(pa + 4 * ks + kOff);

  // Per-lane row metadata (rows rowBase + r + 8*hi).
  float nA[8]; int yA[8];
#pragma unroll
  for (int r = 0; r < 8; ++r) {
    int row = rowBase + r + hi8;
    nA[r] = norms[row];
    yA[r] = y[row & (NHALF - 1)];
  }

  float expsum[8], possum[8], poscnt[8];
#pragma unroll
  for (int r = 0; r < 8; ++r) { expsum[r] = 0.f; possum[r] = 0.f; poscnt[r] = 0.f; }

#if USE_TDM
  if (tid < 32) tdm_load_panel(prow(zi, zj, 0), &Bs[0][0]);  // prologue DMA
#endif

  for (int p = 0; p < NP; ++p) {
    const int cur = p & 1;
    const int jb  = p * BN;

    if (tid < BN) {
      int col = jb + tid;
      nB[cur][tid] = norms[col];
      yB[cur][tid] = y[col & (NHALF - 1)];
    }

#if USE_TDM
    // Wave 0: kick the DMA for the next panel, then wait until the current
    // panel is resident (in-order completion => TENSORcnt<=1 suffices).
    if (tid < 32) {
      if (p + 1 < NP) {
        tdm_load_panel(prow(zi, zj, (p + 1) * BN), &Bs[cur ^ 1][0]);
        __builtin_amdgcn_s_wait_tensorcnt(1);
      } else {
        __builtin_amdgcn_s_wait_tensorcnt(0);
      }
    }
#else
    for (int idx = tid; idx < BN * 32; idx += 256) {
      int r  = idx >> 5;
      int c4 = idx & 31;
      float4 v = *(const float4*)(prow(zi, zj, jb + r) + c4 * 4);
      *(float4*)&Bs[cur][r * LDB + c4 * 4] = v;
    }
#endif
    __syncthreads();

    // --- 4 sub-tiles of 16 columns: 32 chained fp32 WMMAs each -------------
#pragma unroll
    for (int sub = 0; sub < BN / 16; ++sub) {
      const float* bbase = &Bs[cur][(sub * 16 + lo) * LDB + kOff];
      v8f acc = {0.f, 0.f, 0.f, 0.f, 0.f, 0.f, 0.f, 0.f};
#pragma unroll
      for (int ks = 0; ks < 32; ++ks) {
        v2f b = *(const v2f*)(bbase + 4 * ks);
        acc = __builtin_amdgcn_wmma_f32_16x16x4_f32(
            false, areg[ks], false, b, (short)0, acc, false, false);
      }

      // --- fused epilogue: mask, exp, accumulate per row -------------------
      const int   col = jb + sub * 16 + lo;
      const float nj  = nB[cur][sub * 16 + lo];
      const int   yj  = yB[cur][sub * 16 + lo];
#pragma unroll
      for (int r = 0; r < 8; ++r) {
        int   row  = rowBase + r + hi8;
        float sim  = acc[r] / (fmaxf(nA[r] * nj, EPS) * TAU);
        bool  offd = (row != col);
        expsum[r] += offd ? expf(sim) : 0.f;
        if (offd && (yA[r] == yj)) { possum[r] += sim; poscnt[r] += 1.f; }
      }
    }
    __syncthreads();   // all waves done with Bs[cur] before it is re-filled
  }

  // --- reduce across the 16 lanes that share each row ----------------------
#pragma unroll
  for (int r = 0; r < 8; ++r) {
#pragma unroll
    for (int m = 8; m >= 1; m >>= 1) {
      expsum[r] += __shfl_xor(expsum[r], m, 32);
      possum[r] += __shfl_xor(possum[r], m, 32);
      poscnt[r] += __shfl_xor(poscnt[r], m, 32);
    }
  }

  if (lo == 0) {   // lanes 0 and 16 each own 8 distinct rows
#pragma unroll
    for (int r = 0; r < 8; ++r) {
      int row = rowBase + r + hi8;
      mlp[row] = (possum[r] - poscnt[r] * logf(expsum[r])) / poscnt[r];
    }
  }
}

// ----------------- kernel 3: deterministic final reduction -----------------
__global__ void scl_finish_kernel(const float* __restrict__ mlp,
                                  float* __restrict__ out) {
  __shared__ float red[256];
  float s = 0.f;
  for (int i = threadIdx.x; i < N2; i += 256) s += mlp[i];  // fixed order
  red[threadIdx.x] = s;
  __syncthreads();
  for (int m = 128; m > 0; m >>= 1) {
    if (threadIdx.x < m) red[threadIdx.x] += red[threadIdx.x + m];
    __syncthreads();
  }
  if (threadIdx.x == 0) out[0] = -red[0] / (float)N2;
}

// ---------------------------------------------------------------------------
extern "C" void kernel_launch(void* const* d_in, const int* in_sizes, int n_in,
                              void* d_out, int out_size, void* d_ws, size_t ws_size,
                              hipStream_t stream) {
  (void)in_sizes; (void)n_in; (void)out_size; (void)ws_size;
  const float* zi = (const float*)d_in[0];
  const float* zj = (const float*)d_in[1];
  const int*   y  = (const int*)d_in[2];
  float* norms = (float*)d_ws;          // 4096 floats
  float* mlp   = norms + N2;            // 4096 floats
  float* out   = (float*)d_out;

  scl_norms_kernel<<<(N2 * 32) / 256, 256, 0, stream>>>(zi, zj, norms);
  scl_main_kernel<<<N2 / BM, 256, 0, stream>>>(zi, zj, y, norms, mlp);
  scl_finish_kernel<<<1, 256, 0, stream>>>(mlp, out);
}